// AttentionConv_25675314496037
// MI455X (gfx1250) — compile-verified
//
#include <hip/hip_runtime.h>

typedef __attribute__((ext_vector_type(2))) float v2f;
typedef __attribute__((ext_vector_type(8))) float v8f;

#define B_   8
#define C_   64
#define H_   64
#define W_   64
#define O_   64
#define KS   7
#define PAD  3
#define HP   (H_ + 2*PAD)   // 70
#define WP   (W_ + 2*PAD)   // 70
#define SPAD (HP*WP)        // 4900
#define SP   (H_*W_)        // 4096

// ---------------------------------------------------------------------------
// Q GEMM: q[b,o,h,w] = sum_c x[b,c,h,w] * wq[o,c]
// One wave -> 16 consecutive positions (fixed b,h; w0..w0+15) x all 64 outputs.
// Uses V_WMMA_F32_16X16X4_F32 (full fp32 matrix path).
// ---------------------------------------------------------------------------
__global__ __launch_bounds__(128)
void qgemm_kernel(const float* __restrict__ x, const float* __restrict__ wq,
                  float* __restrict__ q) {
    const int lane  = threadIdx.x & 31;
    const int wave  = blockIdx.x * (blockDim.x >> 5) + (threadIdx.x >> 5);
    const int base  = wave * 16;                 // 2048 waves cover 32768 positions
    const int m     = lane & 15;                 // row within tile / col within n-tile
    const int khalf = (lane >> 4) << 1;          // 0 for lanes 0-15, 2 for lanes 16-31

    const int pos = base + m;
    const int b   = pos >> 12;                   // /4096
    const int rem = pos & 4095;
    const float* xb = x + (size_t)b * C_ * SP + rem;   // x[(b*C + c)*4096 + rem]

    v8f acc[4];
    #pragma unroll
    for (int i = 0; i < 4; ++i) acc[i] = (v8f){0,0,0,0,0,0,0,0};

    for (int k0 = 0; k0 < C_; k0 += 4) {
        const int kA = k0 + khalf;
        v2f a;
        a.x = xb[(size_t)kA * SP];               // A: VGPR0 = K0/K2 halves
        a.y = xb[(size_t)(kA + 1) * SP];         //    VGPR1 = K1/K3 halves
        #pragma unroll
        for (int nt = 0; nt < 4; ++nt) {
            const int n = nt * 16 + m;
            v2f bf;
            bf.x = wq[n * C_ + kA];              // B[k,n] = wq[n,k]
            bf.y = wq[n * C_ + kA + 1];
            acc[nt] = __builtin_amdgcn_wmma_f32_16x16x4_f32(
                false, a, false, bf, (short)0, acc[nt], false, false);
        }
    }

    // D layout: VGPR r -> M=r (lanes 0-15) / M=r+8 (lanes 16-31), N=lane%16
    const int bb   = base >> 12;
    const int brem = base & 4095;
    const int hh   = brem >> 6;
    const int w0   = brem & 63;
    #pragma unroll
    for (int rr = 0; rr < 8; ++rr) {
        const int row = (lane < 16) ? rr : rr + 8;
        #pragma unroll
        for (int nt = 0; nt < 4; ++nt) {
            const int o = nt * 16 + m;
            q[(((size_t)bb * O_ + o) * H_ + hh) * W_ + w0 + row] = acc[nt][rr];
        }
    }
}

// ---------------------------------------------------------------------------
// K/V GEMM over the zero-padded 70x70 grid. One A fragment feeds two WMMAs
// (wk and wv) per n-tile per k-step. 39200 positions = exactly 2450 wave tiles.
// ---------------------------------------------------------------------------
__global__ __launch_bounds__(128)
void kvgemm_kernel(const float* __restrict__ x,
                   const float* __restrict__ wk, const float* __restrict__ wv,
                   float* __restrict__ kout, float* __restrict__ vout) {
    const int lane = threadIdx.x & 31;
    const int wave = blockIdx.x * (blockDim.x >> 5) + (threadIdx.x >> 5);
    const int base = wave * 16;
    if (base >= B_ * SPAD) return;               // wave-uniform: EXEC stays full

    const int m     = lane & 15;
    const int khalf = (lane >> 4) << 1;

    const int pos = base + m;
    const int b   = pos / SPAD;
    const int r   = pos - b * SPAD;
    const int hp  = r / WP;
    const int wp  = r - hp * WP;
    const int hs  = hp - PAD;
    const int ws  = wp - PAD;
    const bool valid = (hs >= 0) & (hs < H_) & (ws >= 0) & (ws < W_);
    const float* xb = x + (size_t)b * C_ * SP + hs * W_ + ws;

    v8f acck[4], accv[4];
    #pragma unroll
    for (int i = 0; i < 4; ++i) {
        acck[i] = (v8f){0,0,0,0,0,0,0,0};
        accv[i] = (v8f){0,0,0,0,0,0,0,0};
    }

    for (int k0 = 0; k0 < C_; k0 += 4) {
        const int kA = k0 + khalf;
        v2f a;
        a.x = valid ? xb[(size_t)kA * SP] : 0.0f;        // zero-fill pad region
        a.y = valid ? xb[(size_t)(kA + 1) * SP] : 0.0f;
        #pragma unroll
        for (int nt = 0; nt < 4; ++nt) {
            const int n = nt * 16 + m;
            v2f bk, bv;
            bk.x = wk[n * C_ + kA];  bk.y = wk[n * C_ + kA + 1];
            bv.x = wv[n * C_ + kA];  bv.y = wv[n * C_ + kA + 1];
            acck[nt] = __builtin_amdgcn_wmma_f32_16x16x4_f32(
                false, a, false, bk, (short)0, acck[nt], false, false);
            accv[nt] = __builtin_amdgcn_wmma_f32_16x16x4_f32(
                false, a, false, bv, (short)0, accv[nt], false, false);
        }
    }

    #pragma unroll
    for (int rr = 0; rr < 8; ++rr) {
        const int row  = (lane < 16) ? rr : rr + 8;
        const int posr = base + row;
        const int br   = posr / SPAD;
        const int prr  = posr - br * SPAD;
        #pragma unroll
        for (int nt = 0; nt < 4; ++nt) {
            const int o = nt * 16 + m;
            const size_t addr = ((size_t)br * O_ + o) * SPAD + prr;
            kout[addr] = acck[nt][rr];
            vout[addr] = accv[nt][rr];
        }
    }
}

// ---------------------------------------------------------------------------
// Attention: one workgroup per (b, o, 4-row strip). Stage the 10x70 k/v slabs
// into LDS, then each thread computes one output via 49-tap softmax attention.
// ---------------------------------------------------------------------------
__global__ __launch_bounds__(256)
void attn_kernel(const float* __restrict__ q,
                 const float* __restrict__ kout, const float* __restrict__ vout,
                 const float* __restrict__ rel_w, const float* __restrict__ rel_h,
                 float* __restrict__ out) {
    const int blk   = blockIdx.x;
    const int htile = blk & 15;           // 16 strips of 4 rows
    const int bo    = blk >> 4;           // b*64 + o
    const int o     = bo & 63;
    const int h0    = htile * 4;

    __shared__ float Kl[10 * WP];
    __shared__ float Vl[10 * WP];

    const int tid = threadIdx.x;
    const float* kb = kout + ((size_t)bo * HP + h0) * WP;
    const float* vb = vout + ((size_t)bo * HP + h0) * WP;
    for (int i = tid; i < 10 * WP; i += 256) {   // h0 <= 60, so rows h0..h0+9 < 70
        Kl[i] = kb[i];
        Vl[i] = vb[i];
    }
    __syncthreads();

    const int w  = tid & 63;
    const int hh = tid >> 6;              // 0..3
    const int h  = h0 + hh;

    const float qv   = q[((size_t)bo * H_ + h) * W_ + w];
    const float relv = qv * (rel_w[o * W_ + w] + rel_h[o * H_ + h]);

    float logits[KS * KS];
    float mx = -3.402823466e38f;
    #pragma unroll
    for (int i = 0; i < KS; ++i)
        #pragma unroll
        for (int j = 0; j < KS; ++j) {
            const float l = qv * Kl[(hh + i) * WP + (w + j)] + relv;
            logits[i * KS + j] = l;
            mx = fmaxf(mx, l);
        }

    float s = 0.0f;
    #pragma unroll
    for (int t = 0; t < KS * KS; ++t) {
        const float e = __expf(logits[t] - mx);
        logits[t] = e;
        s += e;
    }
    const float inv = 1.0f / s;

    float accum = 0.0f;
    #pragma unroll
    for (int i = 0; i < KS; ++i)
        #pragma unroll
        for (int j = 0; j < KS; ++j)
            accum += logits[i * KS + j] * Vl[(hh + i) * WP + (w + j)];

    out[((size_t)bo * H_ + h) * W_ + w] = accum * inv;
}

// ---------------------------------------------------------------------------
extern "C" void kernel_launch(void* const* d_in, const int* in_sizes, int n_in,
                              void* d_out, int out_size, void* d_ws, size_t ws_size,
                              hipStream_t stream) {
    const float* x     = (const float*)d_in[0];
    const float* wq    = (const float*)d_in[1];
    const float* wk    = (const float*)d_in[2];
    const float* wv    = (const float*)d_in[3];
    const float* rel_w = (const float*)d_in[4];
    const float* rel_h = (const float*)d_in[5];
    float* out = (float*)d_out;

    // Workspace partition (all fp32):
    //   q    : B*O*H*W        = 2,097,152 floats ( 8.0 MB)
    //   kout : B*O*70*70      = 2,508,800 floats (10.0 MB)
    //   vout : B*O*70*70      = 2,508,800 floats (10.0 MB)
    float* q    = (float*)d_ws;
    float* kout = q    + (size_t)B_ * O_ * SP;
    float* vout = kout + (size_t)B_ * O_ * SPAD;

    // q: 32768 positions / 16 per wave = 2048 waves, 4 waves per block
    qgemm_kernel<<<512, 128, 0, stream>>>(x, wq, q);
    // k/v: 39200 positions / 16 = 2450 waves, 4 per block -> 613 blocks
    kvgemm_kernel<<<613, 128, 0, stream>>>(x, wk, wv, kout, vout);
    // attention: B*O * (H/4) strips = 8192 blocks of 256 threads
    attn_kernel<<<8192, 256, 0, stream>>>(q, kout, vout, rel_w, rel_h, out);
}